// GraphSAGELayer_59596966199955
// MI455X (gfx1250) — compile-verified
//
#include <hip/hip_runtime.h>

// ---------------------------------------------------------------------------
// GraphSAGE layer for MI455X (gfx1250):
//   neigh = segment_sum(x[src], dst)          (L2-resident f32 atomics)
//   out   = neigh @ Wn^T + x @ Ws^T + bn + bs (fused f32 WMMA 16x16x4)
// ---------------------------------------------------------------------------

typedef float v8f __attribute__((ext_vector_type(8)));
typedef float v2f __attribute__((ext_vector_type(2)));

#define IN_DIM  128
#define OUT_DIM 128

// ---------------------------------------------------------------------------
// Kernel 1: zero the neighbor accumulator in workspace (float4 stores).
// ---------------------------------------------------------------------------
__global__ void sage_zero_kernel(float4* __restrict__ p, int n4) {
    int i = blockIdx.x * blockDim.x + threadIdx.x;
    if (i < n4) {
        p[i] = make_float4(0.f, 0.f, 0.f, 0.f);
    }
}

// ---------------------------------------------------------------------------
// Kernel 2: edge scatter-add. One wave32 per edge; each lane moves 4 floats
// (float4 gather -> 4x global_atomic_add_f32). x and neigh both fit in the
// 192MB L2, so this runs at L2/atomic speed after first touch.
// ---------------------------------------------------------------------------
__global__ void sage_scatter_kernel(const float* __restrict__ x,
                                    const long long* __restrict__ edge_index,
                                    float* __restrict__ neigh,
                                    int num_edges) {
    int tid  = blockIdx.x * blockDim.x + threadIdx.x;
    int edge = tid >> 5;          // wave per edge
    int lane = tid & 31;
    if (edge >= num_edges) return;

    long long src = edge_index[edge];                 // row 0: src
    long long dst = edge_index[(long long)num_edges + edge]; // row 1: dst

    const float4 v = *(const float4*)(x + src * IN_DIM + lane * 4);
    float* p = neigh + dst * IN_DIM + lane * 4;
    atomicAdd(p + 0, v.x);
    atomicAdd(p + 1, v.y);
    atomicAdd(p + 2, v.z);
    atomicAdd(p + 3, v.w);
}

// ---------------------------------------------------------------------------
// Kernel 3: fused dual GEMM + bias via V_WMMA_F32_16X16X4_F32.
// One wave per (16-node m-tile, 16-channel o-tile). K=128 -> 32 k-steps,
// 2 WMMAs per step (neigh*Wn and x*Ws share one f32 accumulator).
//
// A 16x4 f32 layout (ISA 7.12.2): lanes 0-15 -> K={k,k+1}, lanes 16-31 ->
// K={k+2,k+3}; one float2 per lane. B mirrored with lane -> output column.
// C/D: VGPR r -> M = r + 8*(lane>=16), N = lane&15.
// ---------------------------------------------------------------------------
__global__ void sage_gemm_kernel(const float* __restrict__ neigh,
                                 const float* __restrict__ x,
                                 const float* __restrict__ W_neigh,
                                 const float* __restrict__ W_self,
                                 const float* __restrict__ b_neigh,
                                 const float* __restrict__ b_self,
                                 float* __restrict__ out,
                                 int num_nodes) {
    const int tid    = blockIdx.x * blockDim.x + threadIdx.x;
    const int wave   = tid >> 5;
    const int lane   = tid & 31;
    const int m_tile = wave >> 3;          // 8 o-tiles of 16 cover OUT_DIM=128
    const int o_tile = wave & 7;
    const int m0     = m_tile * 16;
    const int o0     = o_tile * 16;
    if (m0 >= num_nodes) return;           // wave-uniform; EXEC stays all-ones

    const int row  = lane & 15;            // M (for A) / N-column (for B)
    const int half = lane >> 4;            // K sub-offset selector: +0 or +2

    const float* aN = neigh   + (size_t)(m0 + row) * IN_DIM + half * 2;
    const float* aX = x       + (size_t)(m0 + row) * IN_DIM + half * 2;
    const float* bN = W_neigh + (size_t)(o0 + row) * IN_DIM + half * 2;
    const float* bS = W_self  + (size_t)(o0 + row) * IN_DIM + half * 2;

    v8f acc = {};

#pragma unroll
    for (int k = 0; k < IN_DIM; k += 4) {
        v2f a0 = *(const v2f*)(aN + k);
        v2f b0 = *(const v2f*)(bN + k);
        acc = __builtin_amdgcn_wmma_f32_16x16x4_f32(
            /*neg_a=*/false, a0, /*neg_b=*/false, b0,
            /*c_mod=*/(short)0, acc, /*reuse_a=*/false, /*reuse_b=*/false);

        v2f a1 = *(const v2f*)(aX + k);
        v2f b1 = *(const v2f*)(bS + k);
        acc = __builtin_amdgcn_wmma_f32_16x16x4_f32(
            /*neg_a=*/false, a1, /*neg_b=*/false, b1,
            /*c_mod=*/(short)0, acc, /*reuse_a=*/false, /*reuse_b=*/false);
    }

    const int   col  = lane & 15;
    const float bias = b_neigh[o0 + col] + b_self[o0 + col];

#pragma unroll
    for (int r = 0; r < 8; ++r) {
        const int m = m0 + r + half * 8;
        out[(size_t)m * OUT_DIM + o0 + col] = acc[r] + bias;
    }
}

// ---------------------------------------------------------------------------
// Launch: zero ws accumulator -> scatter-add edges -> WMMA GEMM epilogue.
// ---------------------------------------------------------------------------
extern "C" void kernel_launch(void* const* d_in, const int* in_sizes, int n_in,
                              void* d_out, int out_size, void* d_ws, size_t ws_size,
                              hipStream_t stream) {
    const float*     x   = (const float*)d_in[0];
    const long long* ei  = (const long long*)d_in[1];
    const float*     Wn  = (const float*)d_in[2];
    const float*     bn  = (const float*)d_in[3];
    const float*     Ws  = (const float*)d_in[4];
    const float*     bs  = (const float*)d_in[5];
    float*           out = (float*)d_out;
    float*           nb  = (float*)d_ws;   // [N, 128] f32 accumulator

    const int N = in_sizes[0] / IN_DIM;
    const int E = in_sizes[1] / 2;

    // 1) zero accumulator
    const int n4 = (N * IN_DIM) / 4;
    sage_zero_kernel<<<(n4 + 255) / 256, 256, 0, stream>>>((float4*)nb, n4);

    // 2) scatter-add: one wave32 per edge
    const long long sc_threads = (long long)E * 32;
    sage_scatter_kernel<<<(unsigned)((sc_threads + 255) / 256), 256, 0, stream>>>(
        x, ei, nb, E);

    // 3) fused dual-GEMM + bias: one wave per 16x16 output tile
    const int m_tiles = (N + 15) / 16;
    const long long gm_threads = (long long)m_tiles * 8 /*o-tiles*/ * 32;
    sage_gemm_kernel<<<(unsigned)((gm_threads + 255) / 256), 256, 0, stream>>>(
        nb, x, Wn, Ws, bn, bs, out, N);
}